// RhythmModel_32418413150645
// MI455X (gfx1250) — compile-verified
//
#include <hip/hip_runtime.h>
#include <math.h>

// ---------------------------------------------------------------------------
// Model dims (compile-time constants from the reference)
// ---------------------------------------------------------------------------
#define BATCH 1024
#define SEQ   16
#define EMB   32
#define NHEAD 4
#define HD    8          // head dim = EMB / NHEAD
#define H1    64         // LSTM1 hidden
#define H2    32         // LSTM2 hidden

typedef __attribute__((ext_vector_type(2))) float v2f;
typedef __attribute__((ext_vector_type(8))) float v8f;

__device__ __forceinline__ v8f wmma_f32_16x16x4(v2f a, v2f b, v8f c) {
  // 8 args: (neg_a, A, neg_b, B, c_mod, C, reuse_a, reuse_b)
  return __builtin_amdgcn_wmma_f32_16x16x4_f32(false, a, false, b, (short)0, c,
                                               false, false);
}

__device__ __forceinline__ float sigm(float v) { return 1.0f / (1.0f + __expf(-v)); }

// ---------------------------------------------------------------------------
// Kernel 1: embedding + positional encoding + LSTM1 + LSTM2 + residual.
// One wave per batch element; hidden state shared through LDS.
// Produces zfeat[BATCH][32] = h2_last + h1_last[:32].
// ---------------------------------------------------------------------------
__global__ void __launch_bounds__(256)
k_recurrent(const float* __restrict__ x,
            const float* __restrict__ emb_w, const float* __restrict__ emb_b,
            const float* __restrict__ w1ih, const float* __restrict__ w1hh,
            const float* __restrict__ b1ih, const float* __restrict__ b1hh,
            const float* __restrict__ w2ih, const float* __restrict__ w2hh,
            const float* __restrict__ b2ih, const float* __restrict__ b2hh,
            float* __restrict__ zfeat)
{
  __shared__ float sh[8][128];                 // per wave: xe[32] | h1[64] | h2[32]
  const int wave = threadIdx.x >> 5;
  const int lane = threadIdx.x & 31;
  const int b    = blockIdx.x * 8 + wave;

  float* xe = &sh[wave][0];
  float* h1 = &sh[wave][32];
  float* h2 = &sh[wave][96];

  h1[lane] = 0.0f; h1[lane + 32] = 0.0f; h2[lane] = 0.0f;
  float c1a = 0.0f, c1b = 0.0f, c2 = 0.0f;
  __syncthreads();

  // positional encoding frequency for this feature lane
  const float freq = __expf(-(float)(lane & ~1) * (9.210340371976184f / 32.0f));

  for (int s = 0; s < SEQ; ++s) {
    const float xv  = x[b * SEQ + s];
    const float arg = (float)s * freq;
    const float pe  = ((lane & 1) == 0) ? __sinf(arg) : __cosf(arg);
    xe[lane] = xv * emb_w[lane] + emb_b[lane] + pe;
    __syncthreads();

    // ---- LSTM1: this lane owns hidden units (lane) and (lane+32) ----
    float hn_a = 0.0f, hn_b = 0.0f;
    #pragma unroll 1
    for (int t = 0; t < 2; ++t) {
      const int u = lane + t * 32;
      float g[4];
      #pragma unroll
      for (int gi = 0; gi < 4; ++gi) {
        const int row = gi * H1 + u;
        float acc = b1ih[row] + b1hh[row];
        const float* wi = w1ih + row * 32;
        #pragma unroll 4
        for (int k = 0; k < 32; ++k) acc += wi[k] * xe[k];
        const float* wh = w1hh + row * H1;
        #pragma unroll 4
        for (int k = 0; k < H1; ++k) acc += wh[k] * h1[k];
        g[gi] = acc;
      }
      float& c = t ? c1b : c1a;
      c = sigm(g[1]) * c + sigm(g[0]) * tanhf(g[2]);
      const float hn = sigm(g[3]) * tanhf(c);
      if (t) hn_b = hn; else hn_a = hn;
    }
    __syncthreads();                 // all lanes done reading old h1
    h1[lane] = hn_a; h1[lane + 32] = hn_b;
    __syncthreads();

    // ---- LSTM2: this lane owns hidden unit (lane); input = new h1 ----
    float g2[4];
    #pragma unroll
    for (int gi = 0; gi < 4; ++gi) {
      const int row = gi * H2 + lane;
      float acc = b2ih[row] + b2hh[row];
      const float* wi = w2ih + row * H1;
      #pragma unroll 4
      for (int k = 0; k < H1; ++k) acc += wi[k] * h1[k];
      const float* wh = w2hh + row * H2;
      #pragma unroll 4
      for (int k = 0; k < H2; ++k) acc += wh[k] * h2[k];
      g2[gi] = acc;
    }
    c2 = sigm(g2[1]) * c2 + sigm(g2[0]) * tanhf(g2[2]);
    const float h2n = sigm(g2[3]) * tanhf(c2);
    __syncthreads();                 // all lanes done reading old h2
    h2[lane] = h2n;
    __syncthreads();
  }
  // residual: l2 + l1[:, :, :32] at the last timestep
  zfeat[b * 32 + lane] = h2[lane] + h1[lane];
}

// ---------------------------------------------------------------------------
// Kernel 2: QKV projection  qkv(1024x96) = zfeat(1024x32) @ attn_w.T + attn_b
// One wave per 16x16 output tile, WMMA f32 16x16x4 over K=32 (8 k-steps).
// ---------------------------------------------------------------------------
__global__ void __launch_bounds__(256)
k_qkv(const float* __restrict__ zfeat,
      const float* __restrict__ attn_w, const float* __restrict__ attn_b,
      float* __restrict__ qkv)
{
  const int wave = threadIdx.x >> 5;
  const int lane = threadIdx.x & 31;
  const int wid  = blockIdx.x * 8 + wave;       // 0..383
  const int mt   = wid / 6;                     // row tile (64)
  const int nt   = wid % 6;                     // col tile (6)
  const int half = lane >> 4;
  const int l16  = lane & 15;
  const int col  = nt * 16 + l16;
  const int arow = mt * 16 + l16;

  v8f c;
  const float bias = attn_b[col];
  #pragma unroll
  for (int r = 0; r < 8; ++r) c[r] = bias;

  #pragma unroll
  for (int kk = 0; kk < 8; ++kk) {
    const int k0 = kk * 4 + half * 2;           // ISA A/B f32 operand layout
    v2f a, bm;
    a.x  = zfeat[arow * 32 + k0];
    a.y  = zfeat[arow * 32 + k0 + 1];
    bm.x = attn_w[col * 32 + k0];               // B[k][n] = attn_w[n][k]
    bm.y = attn_w[col * 32 + k0 + 1];
    c = wmma_f32_16x16x4(a, bm, c);
  }
  #pragma unroll
  for (int r = 0; r < 8; ++r)
    qkv[(mt * 16 + r + half * 8) * 96 + col] = c[r];
}

// ---------------------------------------------------------------------------
// Kernel 3: attention over the batch axis (per head), flash-style.
// One wave per (head, 16-query tile).
//   Scores: S^T tile = K_tile(16x8) @ Q^T(8x16)  -> 2 WMMAs per key tile,
//           staged through LDS for the online softmax.
//   P·V:    O(16q x 16d-pad) += P(16x16) @ V(16x8-pad) -> 4 WMMAs per key
//           tile; A operand built directly from the per-lane p[] values
//           (both halves hold identical p for query l16, matching the ISA
//           A layout); per-row rescale factors broadcast via LDS.
// ---------------------------------------------------------------------------
__global__ void __launch_bounds__(256)
k_attn(const float* __restrict__ qkv, float* __restrict__ att)
{
  __shared__ float st[8][16][16];               // per-wave S^T tile
  __shared__ float cb[8][16];                   // per-wave row broadcast
  const int wave = threadIdx.x >> 5;
  const int lane = threadIdx.x & 31;
  const int wid  = blockIdx.x * 8 + wave;       // 0..255
  const int hh   = wid >> 6;                    // head
  const int qt   = wid & 63;                    // query tile
  const int half = lane >> 4;
  const int l16  = lane & 15;
  const float scale = 0.3535533905932738f;      // 1/sqrt(8)

  // B operand (Q^T) loaded once: two k-steps covering d=0..7
  const int qrow  = qt * 16 + l16;
  const int qbase = qrow * 96 + hh * HD;
  v2f bq0, bq1;
  bq0.x = qkv[qbase + half * 2 + 0] * scale;
  bq0.y = qkv[qbase + half * 2 + 1] * scale;
  bq1.x = qkv[qbase + 4 + half * 2 + 0] * scale;
  bq1.y = qkv[qbase + 4 + half * 2 + 1] * scale;

  // per-lane softmax state for query l16 (duplicated across halves)
  float mx = -1e30f, sum = 0.0f;
  // O accumulator in D layout: row = query (r + half*8), col = value dim l16
  v8f oc = {0.0f, 0.0f, 0.0f, 0.0f, 0.0f, 0.0f, 0.0f, 0.0f};
  const int vcol = 64 + hh * HD + l16;          // V column this lane supplies

  #pragma unroll 1
  for (int mt = 0; mt < 64; ++mt) {
    // ---- scores: A operand = K tile (16 keys x 8) ----
    const int krow  = mt * 16 + l16;
    const int kbase = krow * 96 + 32 + hh * HD;
    v2f a0, a1;
    a0.x = qkv[kbase + half * 2 + 0];
    a0.y = qkv[kbase + half * 2 + 1];
    a1.x = qkv[kbase + 4 + half * 2 + 0];
    a1.y = qkv[kbase + 4 + half * 2 + 1];
    if (mt + 1 < 64)
      __builtin_prefetch(qkv + (mt * 16 + 16 + l16) * 96 + 32 + hh * HD, 0, 1);

    v8f c = {0.0f, 0.0f, 0.0f, 0.0f, 0.0f, 0.0f, 0.0f, 0.0f};
    c = wmma_f32_16x16x4(a0, bq0, c);
    c = wmma_f32_16x16x4(a1, bq1, c);

    __syncthreads();                            // prev-iter LDS reads done
    #pragma unroll
    for (int r = 0; r < 8; ++r)
      st[wave][r + half * 8][l16] = c[r];       // S^T: [key][query]
    __syncthreads();

    // ---- online softmax over this key tile for query l16 ----
    float tmax = mx;
    float p[16];
    #pragma unroll
    for (int m = 0; m < 16; ++m) tmax = fmaxf(tmax, st[wave][m][l16]);
    const float corr = __expf(mx - tmax);
    mx = tmax;
    float psum = 0.0f;
    #pragma unroll
    for (int m = 0; m < 16; ++m) { p[m] = __expf(st[wave][m][l16] - mx); psum += p[m]; }
    sum = sum * corr + psum;

    // broadcast per-query correction factor for the D-layout rescale
    if (half == 0) cb[wave][l16] = corr;
    __syncthreads();
    #pragma unroll
    for (int r = 0; r < 8; ++r) oc[r] *= cb[wave][r + half * 8];

    // ---- P·V via WMMA: A = P (per-lane p[] is exactly the A layout),
    //      B = V tile (head dim 8 zero-padded to 16 columns) ----
    #pragma unroll
    for (int kk = 0; kk < 4; ++kk) {
      const int k0 = kk * 4 + half * 2;
      v2f a, bv;
      a.x = p[k0];
      a.y = p[k0 + 1];
      if (l16 < HD) {
        bv.x = qkv[(mt * 16 + k0 + 0) * 96 + vcol];
        bv.y = qkv[(mt * 16 + k0 + 1) * 96 + vcol];
      } else {
        bv.x = 0.0f; bv.y = 0.0f;
      }
      oc = wmma_f32_16x16x4(a, bv, oc);
    }
  }

  // broadcast per-query softmax denominator, then normalize + store
  if (half == 0) cb[wave][l16] = sum;
  __syncthreads();
  if (l16 < HD) {
    #pragma unroll
    for (int r = 0; r < 8; ++r) {
      const int q = r + half * 8;
      att[(qt * 16 + q) * 32 + hh * HD + l16] = oc[r] / cb[wave][q];
    }
  }
}

// ---------------------------------------------------------------------------
// Kernel 4: output projection + d1/bn1/relu + d2/bn2/relu + d3/sigmoid.
// One wave per 16-row tile; chained WMMA GEMMs with LDS restaging.
// ---------------------------------------------------------------------------
__device__ __forceinline__ v8f mm_tile16(const float* __restrict__ Abuf, int sA,
                                         const float* __restrict__ W, int Kin,
                                         int col, int half, int l16, float cinit)
{
  v8f c;
  #pragma unroll
  for (int r = 0; r < 8; ++r) c[r] = cinit;
  #pragma unroll 4
  for (int kk = 0; kk < Kin / 4; ++kk) {
    const int k0 = kk * 4 + half * 2;
    v2f a, bm;
    a.x  = Abuf[l16 * sA + k0];
    a.y  = Abuf[l16 * sA + k0 + 1];
    bm.x = W[col * Kin + k0];
    bm.y = W[col * Kin + k0 + 1];
    c = wmma_f32_16x16x4(a, bm, c);
  }
  return c;
}

__global__ void __launch_bounds__(128)
k_head(const float* __restrict__ att,
       const float* __restrict__ ow,  const float* __restrict__ ob,
       const float* __restrict__ d1w, const float* __restrict__ d1b,
       const float* __restrict__ g1,  const float* __restrict__ bb1,
       const float* __restrict__ d2w, const float* __restrict__ d2b,
       const float* __restrict__ g2,  const float* __restrict__ bb2,
       const float* __restrict__ d3w, const float* __restrict__ d3b,
       float* __restrict__ out)
{
  __shared__ float bufA[4][16][64];
  __shared__ float bufB[4][16][64];
  const int wave = threadIdx.x >> 5;
  const int lane = threadIdx.x & 31;
  const int rt   = blockIdx.x * 4 + wave;       // 16-row tile, 0..63
  const int half = lane >> 4;
  const int l16  = lane & 15;
  const float invs = 0.9999950000374997f;       // 1/sqrt(1+1e-5)

  // stage the att tile (16x32) into bufA
  for (int i = lane; i < 512; i += 32) {
    const int r = i >> 5, cc = i & 31;
    bufA[wave][r][cc] = att[(rt * 16 + r) * 32 + cc];
  }
  __syncthreads();

  // L0: z0 = att @ ow^T + ob   (16x32)
  #pragma unroll
  for (int nt = 0; nt < 2; ++nt) {
    const int col = nt * 16 + l16;
    v8f c = mm_tile16(&bufA[wave][0][0], 64, ow, 32, col, half, l16, ob[col]);
    #pragma unroll
    for (int r = 0; r < 8; ++r) bufB[wave][r + half * 8][col] = c[r];
  }
  __syncthreads();

  // L1: z1 = relu(bn1(z0 @ d1w^T + d1b))   (16x64)
  #pragma unroll
  for (int nt = 0; nt < 4; ++nt) {
    const int col = nt * 16 + l16;
    v8f c = mm_tile16(&bufB[wave][0][0], 64, d1w, 32, col, half, l16, d1b[col]);
    const float s = g1[col] * invs, bsh = bb1[col];
    #pragma unroll
    for (int r = 0; r < 8; ++r)
      bufA[wave][r + half * 8][col] = fmaxf(c[r] * s + bsh, 0.0f);
  }
  __syncthreads();

  // L2: z2 = relu(bn2(z1 @ d2w^T + d2b))   (16x32)
  #pragma unroll
  for (int nt = 0; nt < 2; ++nt) {
    const int col = nt * 16 + l16;
    v8f c = mm_tile16(&bufA[wave][0][0], 64, d2w, 64, col, half, l16, d2b[col]);
    const float s = g2[col] * invs, bsh = bb2[col];
    #pragma unroll
    for (int r = 0; r < 8; ++r)
      bufB[wave][r + half * 8][col] = fmaxf(c[r] * s + bsh, 0.0f);
  }
  __syncthreads();

  // L3: out = sigmoid(z2 @ d3w^T + d3b)   (16x1) -- per-lane dot
  if (half == 0) {
    float acc = d3b[0];
    #pragma unroll
    for (int k = 0; k < 32; ++k) acc += bufB[wave][l16][k] * d3w[k];
    out[rt * 16 + l16] = 1.0f / (1.0f + __expf(-acc));
  }
}

// ---------------------------------------------------------------------------
// Host-side launcher
// ---------------------------------------------------------------------------
extern "C" void kernel_launch(void* const* d_in, const int* in_sizes, int n_in,
                              void* d_out, int out_size, void* d_ws, size_t ws_size,
                              hipStream_t stream) {
  const float* x      = (const float*)d_in[0];
  const float* emb_w  = (const float*)d_in[1];
  const float* emb_b  = (const float*)d_in[2];
  const float* l1_wih = (const float*)d_in[3];
  const float* l1_whh = (const float*)d_in[4];
  const float* l1_bih = (const float*)d_in[5];
  const float* l1_bhh = (const float*)d_in[6];
  const float* l2_wih = (const float*)d_in[7];
  const float* l2_whh = (const float*)d_in[8];
  const float* l2_bih = (const float*)d_in[9];
  const float* l2_bhh = (const float*)d_in[10];
  const float* attn_w = (const float*)d_in[11];
  const float* attn_b = (const float*)d_in[12];
  const float* attn_ow= (const float*)d_in[13];
  const float* attn_ob= (const float*)d_in[14];
  const float* d1_w   = (const float*)d_in[15];
  const float* d1_b   = (const float*)d_in[16];
  const float* bn1_g  = (const float*)d_in[17];
  const float* bn1_b  = (const float*)d_in[18];
  const float* d2_w   = (const float*)d_in[19];
  const float* d2_b   = (const float*)d_in[20];
  const float* bn2_g  = (const float*)d_in[21];
  const float* bn2_b  = (const float*)d_in[22];
  const float* d3_w   = (const float*)d_in[23];
  const float* d3_b   = (const float*)d_in[24];
  float* out = (float*)d_out;

  float* zfeat = (float*)d_ws;                  // 1024*32
  float* qkv   = zfeat + BATCH * 32;            // 1024*96
  float* att   = qkv   + BATCH * 96;            // 1024*32

  k_recurrent<<<BATCH / 8, 256, 0, stream>>>(x, emb_w, emb_b,
                                             l1_wih, l1_whh, l1_bih, l1_bhh,
                                             l2_wih, l2_whh, l2_bih, l2_bhh,
                                             zfeat);
  k_qkv<<<48, 256, 0, stream>>>(zfeat, attn_w, attn_b, qkv);
  k_attn<<<32, 256, 0, stream>>>(qkv, att);
  k_head<<<16, 128, 0, stream>>>(att, attn_ow, attn_ob,
                                 d1_w, d1_b, bn1_g, bn1_b,
                                 d2_w, d2_b, bn2_g, bn2_b,
                                 d3_w, d3_b, out);
}